// Weighted3DTransformLoss_23502061044191
// MI455X (gfx1250) — compile-verified
//
#include <hip/hip_runtime.h>

typedef float v2f __attribute__((ext_vector_type(2)));
typedef float v8f __attribute__((ext_vector_type(8)));

#define HW_PIX (240 * 320)
#define NK 8
#define NB 64

// pred[b,i,pix] = sum_k M_bk(3x4) @ (mask_k * [x,y,z,1])  ; loss = 0.5*sum(diff^2)/size
// One wave processes 16 pixels per iteration via V_WMMA_F32_16X16X4_F32 chained over k.
__global__ void __launch_bounds__(256)
w3dt_loss_wmma(const float* __restrict__ pts,
               const float* __restrict__ masks,
               const float* __restrict__ tfms,
               const float* __restrict__ tgt,
               float* __restrict__ out)
{
    const int b          = blockIdx.x;
    const int lane       = threadIdx.x & 31;
    const int waveInBlk  = threadIdx.x >> 5;
    const int wavesPerBl = blockDim.x >> 5;
    const int half       = lane >> 4;   // 0: K=0,1 ; 1: K=2,3
    const int l          = lane & 15;   // A: row M ; B: pixel column N

    // ---- Hoist A matrices: [R|t] (3x4) padded to 16x4, one per k ----
    // lane<16 : (M=l, K=0),(M=l, K=1)  -> R[l][0], R[l][1]
    // lane>=16: (M=l, K=2),(M=l, K=3)  -> R[l][2], t[l]
    v2f A[NK];
#pragma unroll
    for (int k = 0; k < NK; ++k) {
        v2f a = {0.f, 0.f};
        if (l < 3) {
            const float* base = tfms + (((b * NK + k) * 3 + l) << 2) + (half ? 2 : 0);
            a.x = base[0];
            a.y = base[1];
        }
        A[k] = a;
    }

    const float* ptsB = pts   + (size_t)b * 3  * HW_PIX;
    const float* tgtB = tgt   + (size_t)b * 3  * HW_PIX;
    const float* mskB = masks + (size_t)b * NK * HW_PIX;

    const int G      = HW_PIX / 16;                      // 4800 pixel-groups per batch
    const int wave   = blockIdx.y * wavesPerBl + waveInBlk;
    const int stride = gridDim.y * wavesPerBl;

    float acc = 0.f;
    const float live = (half == 0) ? 1.f : 0.f;          // only lanes 0..15 hold rows 0..2

    for (int g = wave; g < G; g += stride) {             // uniform trip count per wave
        const int p = g * 16 + l;

        const float x = ptsB[p];
        const float y = ptsB[HW_PIX + p];
        const float z = ptsB[2 * HW_PIX + p];
        const float q0 = half ? z : x;                   // B row pair this half-wave owns
        const float q1 = half ? 1.f : y;

        v8f c = {};                                      // C/D accumulator, chained over k
#pragma unroll
        for (int k = 0; k < NK; ++k) {
            const float m = mskB[(size_t)k * HW_PIX + p];
            v2f bb;
            bb.x = m * q0;
            bb.y = m * q1;
            c = __builtin_amdgcn_wmma_f32_16x16x4_f32(
                    /*neg_a=*/false, A[k],
                    /*neg_b=*/false, bb,
                    /*c_mod=*/(short)0, c,
                    /*reuse_a=*/false, /*reuse_b=*/false);
        }

        const float tx = tgtB[p];
        const float ty = tgtB[HW_PIX + p];
        const float tz = tgtB[2 * HW_PIX + p];
        const float d0 = c[0] - tx;                      // M=0 row (lanes 0..15)
        const float d1 = c[1] - ty;                      // M=1 row
        const float d2 = c[2] - tz;                      // M=2 row
        acc += live * (d0 * d0 + d1 * d1 + d2 * d2);
    }

    // ---- wave32 reduction, then one global atomic per wave ----
#pragma unroll
    for (int off = 16; off > 0; off >>= 1)
        acc += __shfl_xor(acc, off, 32);

    if (lane == 0) {
        const float scale = 0.5f / (float)((size_t)NB * 3 * HW_PIX); // SIZE_AVERAGE
        atomicAdd(out, acc * scale);
    }
}

extern "C" void kernel_launch(void* const* d_in, const int* in_sizes, int n_in,
                              void* d_out, int out_size, void* d_ws, size_t ws_size,
                              hipStream_t stream) {
    (void)in_sizes; (void)n_in; (void)out_size; (void)d_ws; (void)ws_size;
    const float* pts   = (const float*)d_in[0];
    const float* masks = (const float*)d_in[1];
    const float* tfms  = (const float*)d_in[2];
    const float* tgt   = (const float*)d_in[3];
    float* out = (float*)d_out;

    hipMemsetAsync(out, 0, sizeof(float), stream);       // graph-capture-safe zero

    dim3 grid(NB, 75);   // 64 batches x 75 chunks -> 4800 blocks, 8 wave32 each
    dim3 block(256);
    w3dt_loss_wmma<<<grid, block, 0, stream>>>(pts, masks, tfms, tgt, out);
}